// HeterogeneousConvoGNN_31250182045924
// MI455X (gfx1250) — compile-verified
//
#include <hip/hip_runtime.h>
#include <math.h>

#define HID 128
#define NHEAD 4
#define DDIM 32

typedef float v2f __attribute__((ext_vector_type(2)));
typedef float v8f __attribute__((ext_vector_type(8)));

#define ENC_NEG_INF 0x007FFFFFu
#define INV_SQRT_D 0.17677669529663687f

__device__ __forceinline__ float gelu_exact(float x) {
    return 0.5f * x * (1.0f + erff(x * 0.7071067811865475f));
}
__device__ __forceinline__ unsigned enc_f(float f) {
    unsigned u = __float_as_uint(f);
    return (u & 0x80000000u) ? ~u : (u | 0x80000000u);
}
__device__ __forceinline__ float dec_f(unsigned k) {
    return (k & 0x80000000u) ? __uint_as_float(k & 0x7fffffffu)
                             : __uint_as_float(~k);
}

// ---------------------------------------------------------------------------
// Dense GEMM: C[N,128] = act(A)[N,128] @ W[128,128] + bias, optional skip mix.
// One wave -> one 16-row x 128-col strip (8 f32 WMMA accumulators, 256 WMMAs).
// W staged in LDS pair-interleaved: sW[kpair][col] = (W[2k][col], W[2k+1][col])
// so each lane's B fragment is a single aligned ds_load_b64 (no repacking).
// Epilogue: wave-uniform bounds test -> fast path is 64 stores off one base
// address with immediate offsets (no per-element exec manipulation).
// ---------------------------------------------------------------------------
template<int GELU_IN, int COMBINE>
__global__ __launch_bounds__(256) void gemm128(
    const float* __restrict__ A, const float* __restrict__ W,
    const float* __restrict__ bias, const float* __restrict__ hprev,
    const float* __restrict__ skipPtr, float* __restrict__ C, int N)
{
    __shared__ v2f sW[(HID / 2) * HID];          // 64KB
    for (int i = threadIdx.x; i < (HID / 2) * HID; i += 256) {
        const int kp = i >> 7, col = i & 127;
        v2f w2;
        w2.x = W[(2 * kp)     * HID + col];
        w2.y = W[(2 * kp + 1) * HID + col];
        sW[i] = w2;
    }
    __syncthreads();

    const int wave  = threadIdx.x >> 5;
    const int lane  = threadIdx.x & 31;
    const int strip = blockIdx.x * 8 + wave;
    const int row0  = strip * 16;
    if (row0 >= N) return;                 // wave-uniform exit, EXEC stays full

    const int M    = lane & 15;            // A row within strip (fp32 A 16x4 layout)
    const int koff = (lane >> 4) << 1;     // lanes 16-31 hold K+2,K+3
    const int nn   = lane & 15;            // output column within 16-tile
    int arow = row0 + M; if (arow >= N) arow = N - 1;
    const float* Arow = A + (size_t)arow * HID;

    v8f zero = {0.f,0.f,0.f,0.f,0.f,0.f,0.f,0.f};
    v8f acc[8];
    #pragma unroll
    for (int t = 0; t < 8; ++t) acc[t] = zero;

    for (int k0 = 0; k0 < HID; k0 += 4) {
        v2f a = *(const v2f*)(Arow + k0 + koff);
        if (GELU_IN) { a.x = gelu_exact(a.x); a.y = gelu_exact(a.y); }
        const v2f* wp = sW + ((k0 + koff) >> 1) * HID;
        #pragma unroll
        for (int t = 0; t < 8; ++t) {
            const v2f b = wp[t * 16 + nn];
            acc[t] = __builtin_amdgcn_wmma_f32_16x16x4_f32(
                false, a, false, b, (short)0, acc[t], false, false);
        }
    }

    float aS = 0.f;
    if (COMBINE) aS = 1.f / (1.f + __expf(-skipPtr[0]));
    const int rbase = (lane >> 4) * 8;     // C layout: lanes>=16 hold M=r+8
    float bb[8];
    #pragma unroll
    for (int t = 0; t < 8; ++t) bb[t] = bias ? bias[t * 16 + nn] : 0.f;

    if (row0 + 16 <= N) {
        // fast path: one base address, all offsets are immediates
        float* Cb = C + (size_t)(row0 + rbase) * HID + nn;
        const float* Hb = COMBINE ? hprev + (size_t)(row0 + rbase) * HID + nn
                                  : (const float*)nullptr;
        #pragma unroll
        for (int t = 0; t < 8; ++t) {
            #pragma unroll
            for (int r = 0; r < 8; ++r) {
                float o = acc[t][r] + bb[t];
                if (COMBINE)
                    o = aS * o + (1.f - aS) * Hb[r * HID + t * 16];
                Cb[r * HID + t * 16] = o;
            }
        }
    } else {
        #pragma unroll
        for (int t = 0; t < 8; ++t) {
            const int col = t * 16 + nn;
            #pragma unroll
            for (int r = 0; r < 8; ++r) {
                const int row = row0 + rbase + r;
                if (row < N) {
                    float o = acc[t][r] + bb[t];
                    if (COMBINE)
                        o = aS * o + (1.f - aS) * hprev[(size_t)row * HID + col];
                    C[(size_t)row * HID + col] = o;
                }
            }
        }
    }
}

// ---------------------------------------------------------------------------
// Per-head relation transform: C[n, h*32 + :] = A[n, h*32 + :] @ Wrel[h] (32x32)
// grid.y = head. One wave -> 16 rows x 32 cols (2 WMMA tiles, K=32).
// ---------------------------------------------------------------------------
__global__ __launch_bounds__(256) void rel_gemm(
    const float* __restrict__ A, const float* __restrict__ Wrel,
    float* __restrict__ C, int N)
{
    const int h = blockIdx.y;
    __shared__ v2f sW[(DDIM / 2) * DDIM];
    const float* Wh = Wrel + h * DDIM * DDIM;
    for (int i = threadIdx.x; i < (DDIM / 2) * DDIM; i += 256) {
        const int kp = i >> 5, col = i & 31;
        v2f w2;
        w2.x = Wh[(2 * kp)     * DDIM + col];
        w2.y = Wh[(2 * kp + 1) * DDIM + col];
        sW[i] = w2;
    }
    __syncthreads();

    const int wave  = threadIdx.x >> 5;
    const int lane  = threadIdx.x & 31;
    const int strip = blockIdx.x * 8 + wave;
    const int row0  = strip * 16;
    if (row0 >= N) return;

    const int M    = lane & 15;
    const int koff = (lane >> 4) << 1;
    const int nn   = lane & 15;
    int arow = row0 + M; if (arow >= N) arow = N - 1;
    const float* Arow = A + (size_t)arow * HID + h * DDIM;

    v8f zero = {0.f,0.f,0.f,0.f,0.f,0.f,0.f,0.f};
    v8f acc[2] = { zero, zero };

    for (int k0 = 0; k0 < DDIM; k0 += 4) {
        v2f a = *(const v2f*)(Arow + k0 + koff);
        const v2f* wp = sW + ((k0 + koff) >> 1) * DDIM;
        #pragma unroll
        for (int t = 0; t < 2; ++t) {
            const v2f b = wp[t * 16 + nn];
            acc[t] = __builtin_amdgcn_wmma_f32_16x16x4_f32(
                false, a, false, b, (short)0, acc[t], false, false);
        }
    }

    const int rbase = (lane >> 4) * 8;
    if (row0 + 16 <= N) {
        float* Cb = C + (size_t)(row0 + rbase) * HID + h * DDIM + nn;
        #pragma unroll
        for (int t = 0; t < 2; ++t)
            #pragma unroll
            for (int r = 0; r < 8; ++r)
                Cb[r * HID + t * 16] = acc[t][r];
    } else {
        #pragma unroll
        for (int t = 0; t < 2; ++t)
            #pragma unroll
            for (int r = 0; r < 8; ++r) {
                const int row = row0 + rbase + r;
                if (row < N)
                    C[(size_t)row * HID + h * DDIM + t * 16 + nn] = acc[t][r];
            }
    }
}

// ---------------------------------------------------------------------------
// Edge pass A: per-edge logits (wave per edge, lane = d), atomic segment max.
// ---------------------------------------------------------------------------
__global__ __launch_bounds__(256) void edge_logits(
    const int* __restrict__ src, const int* __restrict__ dst,
    const float* __restrict__ q, const float* __restrict__ ke,
    const float* __restrict__ prel, float* __restrict__ lg,
    unsigned* __restrict__ mx, int E)
{
    const int e    = (int)((blockIdx.x * 256 + threadIdx.x) >> 5);
    const int lane = threadIdx.x & 31;
    if (e >= E) return;
    const int s = src[e], d = dst[e];
    const float* qd = q  + (size_t)d * HID;
    const float* ks = ke + (size_t)s * HID;
    #pragma unroll
    for (int h = 0; h < NHEAD; ++h) {
        float p = qd[h * DDIM + lane] * ks[h * DDIM + lane];
        #pragma unroll
        for (int off = 16; off > 0; off >>= 1) p += __shfl_xor(p, off, 32);
        if (lane == 0) {
            const float L = p * prel[h] * INV_SQRT_D;
            lg[(size_t)e * NHEAD + h] = L;
            atomicMax(&mx[(size_t)d * NHEAD + h], enc_f(L));
        }
    }
}

// ---------------------------------------------------------------------------
// Edge pass B: w = exp(lg - mx[dst]); scatter-add w and w*msg.
// ---------------------------------------------------------------------------
__global__ __launch_bounds__(256) void edge_agg(
    const int* __restrict__ src, const int* __restrict__ dst,
    const float* __restrict__ lg, const unsigned* __restrict__ mx,
    const float* __restrict__ me, float* __restrict__ den,
    float* __restrict__ agg, int E)
{
    const int e    = (int)((blockIdx.x * 256 + threadIdx.x) >> 5);
    const int lane = threadIdx.x & 31;
    if (e >= E) return;
    const int s = src[e], d = dst[e];
    const float* ms = me  + (size_t)s * HID;
    float*       ad = agg + (size_t)d * HID;
    #pragma unroll
    for (int h = 0; h < NHEAD; ++h) {
        const float m = dec_f(mx[(size_t)d * NHEAD + h]);
        const float w = __expf(lg[(size_t)e * NHEAD + h] - m);
        if (lane == 0) atomicAdd(&den[(size_t)d * NHEAD + h], w);
        atomicAdd(&ad[h * DDIM + lane], w * ms[h * DDIM + lane]);
    }
}

// agg[i] /= den  (float4 per thread; 8 float4 per head-block of 32)
__global__ void normalize_agg(float4* __restrict__ agg,
                              const float* __restrict__ den, size_t n32)
{
    size_t i = (size_t)blockIdx.x * blockDim.x + threadIdx.x;
    if (i >= n32) return;
    const size_t node = i >> 5;
    const int h = (int)((i >> 3) & 3);
    const float dn = den[node * NHEAD + h];
    const float r = (dn > 0.f) ? 1.f / dn : 0.f;
    float4 v = agg[i];
    v.x *= r; v.y *= r; v.z *= r; v.w *= r;
    agg[i] = v;
}

// ---------------------------------------------------------------------------
// y = LayerNorm(gelu(x)) * g + b — one wave per row, float4 per lane (b128).
// ---------------------------------------------------------------------------
__global__ __launch_bounds__(256) void gelu_ln(
    const float* __restrict__ X, const float* __restrict__ g,
    const float* __restrict__ b, float* __restrict__ Y, int N)
{
    const int row  = (int)((blockIdx.x * 256 + threadIdx.x) >> 5);
    const int lane = threadIdx.x & 31;
    if (row >= N) return;
    const float4 xin = ((const float4*)(X + (size_t)row * HID))[lane];
    float x[4] = { gelu_exact(xin.x), gelu_exact(xin.y),
                   gelu_exact(xin.z), gelu_exact(xin.w) };
    float s = x[0] + x[1] + x[2] + x[3];
    #pragma unroll
    for (int off = 16; off > 0; off >>= 1) s += __shfl_xor(s, off, 32);
    const float mu = s * (1.f / 128.f);
    float vs = 0.f;
    #pragma unroll
    for (int j = 0; j < 4; ++j) { const float dd = x[j] - mu; vs += dd * dd; }
    #pragma unroll
    for (int off = 16; off > 0; off >>= 1) vs += __shfl_xor(vs, off, 32);
    const float inv = rsqrtf(vs * (1.f / 128.f) + 1e-5f);
    const float4 gg = ((const float4*)g)[lane];
    const float4 bb = ((const float4*)b)[lane];
    float4 o;
    o.x = (x[0] - mu) * inv * gg.x + bb.x;
    o.y = (x[1] - mu) * inv * gg.y + bb.y;
    o.z = (x[2] - mu) * inv * gg.z + bb.z;
    o.w = (x[3] - mu) * inv * gg.w + bb.w;
    ((float4*)(Y + (size_t)row * HID))[lane] = o;
}

__global__ void fill_u32(unsigned* __restrict__ p, unsigned v, size_t n) {
    size_t i = (size_t)blockIdx.x * blockDim.x + threadIdx.x;
    const size_t stride = (size_t)gridDim.x * blockDim.x;
    for (; i < n; i += stride) p[i] = v;
}

// ---------------------------------------------------------------------------
extern "C" void kernel_launch(void* const* d_in, const int* in_sizes, int n_in,
                              void* d_out, int out_size, void* d_ws, size_t ws_size,
                              hipStream_t stream)
{
    (void)n_in; (void)out_size; (void)ws_size;
    const float* x_utt = (const float*)d_in[0];
    const float* x_spk = (const float*)d_in[1];
    const float* Wk   = (const float*)d_in[2];
    const float* bk   = (const float*)d_in[3];
    const float* Wq   = (const float*)d_in[4];
    const float* bq   = (const float*)d_in[5];
    const float* Wv   = (const float*)d_in[6];
    const float* bv   = (const float*)d_in[7];
    const float* a_rel = (const float*)d_in[8];
    const float* m_rel = (const float*)d_in[9];
    const float* p_rel = (const float*)d_in[10];
    const float* Wa   = (const float*)d_in[11];
    const float* ba   = (const float*)d_in[12];
    const float* skip = (const float*)d_in[13];
    const float* ln_g = (const float*)d_in[14];
    const float* ln_b = (const float*)d_in[15];
    const int* srcE[3] = { (const int*)d_in[16], (const int*)d_in[18], (const int*)d_in[20] };
    const int* dstE[3] = { (const int*)d_in[17], (const int*)d_in[19], (const int*)d_in[21] };

    const int Nu = in_sizes[0] / HID;
    const int Ns = in_sizes[1] / HID;
    const int Nt[2]   = { Nu, Ns };
    const int Ecnt[3] = { in_sizes[16], in_sizes[18], in_sizes[20] };
    const int ET_SRC[3] = { 0, 0, 1 };
    const int ET_DST[3] = { 0, 1, 0 };

    float* p = (float*)d_ws;
    auto alloc = [&](size_t n) { float* r = p; p += n; return r; };
    float* h1[2]   = { alloc((size_t)Nu * HID), alloc((size_t)Ns * HID) };
    float* kbuf[2] = { alloc((size_t)Nu * HID), alloc((size_t)Ns * HID) };
    float* qbuf[2] = { alloc((size_t)Nu * HID), alloc((size_t)Ns * HID) };
    float* vbuf[2] = { alloc((size_t)Nu * HID), alloc((size_t)Ns * HID) };
    float* kearr[3]; float* mearr[3];
    for (int e = 0; e < 3; ++e) {
        const size_t n = (size_t)Nt[ET_SRC[e]] * HID;
        kearr[e] = alloc(n); mearr[e] = alloc(n);
    }
    float* lg[3];
    for (int e = 0; e < 3; ++e) lg[e] = alloc((size_t)Ecnt[e] * NHEAD);
    unsigned* mx[2] = { (unsigned*)alloc((size_t)Nu * NHEAD),
                        (unsigned*)alloc((size_t)Ns * NHEAD) };
    float* den[2] = { alloc((size_t)Nu * NHEAD), alloc((size_t)Ns * NHEAD) };
    float* agg[2] = { alloc((size_t)Nu * HID),  alloc((size_t)Ns * HID)  };
    float* opre[2] = { alloc((size_t)Nu * HID), alloc((size_t)Ns * HID) };

    const float* hin[2] = { x_utt, x_spk };

    for (int l = 0; l < 2; ++l) {
        // ---- K/Q/V projections (WMMA GEMMs) ----
        for (int t = 0; t < 2; ++t) {
            const int N = Nt[t];
            const int strips = (N + 15) / 16;
            dim3 grid((strips + 7) / 8);
            const size_t woff = (size_t)(l * 2 + t) * HID * HID;
            const size_t boff = (size_t)(l * 2 + t) * HID;
            gemm128<0,0><<<grid, 256, 0, stream>>>(hin[t], Wk + woff, bk + boff,
                                                   nullptr, nullptr, kbuf[t], N);
            gemm128<0,0><<<grid, 256, 0, stream>>>(hin[t], Wq + woff, bq + boff,
                                                   nullptr, nullptr, qbuf[t], N);
            gemm128<0,0><<<grid, 256, 0, stream>>>(hin[t], Wv + woff, bv + boff,
                                                   nullptr, nullptr, vbuf[t], N);
        }
        // ---- per-relation head transforms (WMMA, K=32) ----
        for (int e = 0; e < 3; ++e) {
            const int s = ET_SRC[e];
            const int N = Nt[s];
            const int strips = (N + 15) / 16;
            dim3 grid((strips + 7) / 8, NHEAD);
            const size_t roff = (size_t)(l * 3 + e) * NHEAD * DDIM * DDIM;
            rel_gemm<<<grid, 256, 0, stream>>>(kbuf[s], a_rel + roff, kearr[e], N);
            rel_gemm<<<grid, 256, 0, stream>>>(vbuf[s], m_rel + roff, mearr[e], N);
        }
        // ---- init softmax accumulators ----
        for (int t = 0; t < 2; ++t) {
            const size_t n4 = (size_t)Nt[t] * NHEAD;
            const size_t n128 = (size_t)Nt[t] * HID;
            fill_u32<<<256, 256, 0, stream>>>(mx[t], ENC_NEG_INF, n4);
            fill_u32<<<256, 256, 0, stream>>>((unsigned*)den[t], 0u, n4);
            fill_u32<<<256, 256, 0, stream>>>((unsigned*)agg[t], 0u, n128);
        }
        // ---- edge pass A: logits + segment max ----
        for (int e = 0; e < 3; ++e) {
            const int dt = ET_DST[e];
            dim3 grid((Ecnt[e] + 7) / 8);
            edge_logits<<<grid, 256, 0, stream>>>(srcE[e], dstE[e], qbuf[dt],
                kearr[e], p_rel + (size_t)(l * 3 + e) * NHEAD, lg[e], mx[dt], Ecnt[e]);
        }
        // ---- edge pass B: exp + scatter add ----
        for (int e = 0; e < 3; ++e) {
            const int dt = ET_DST[e];
            dim3 grid((Ecnt[e] + 7) / 8);
            edge_agg<<<grid, 256, 0, stream>>>(srcE[e], dstE[e], lg[e], mx[dt],
                mearr[e], den[dt], agg[dt], Ecnt[e]);
        }
        // ---- normalize ----
        for (int t = 0; t < 2; ++t) {
            const size_t n32 = (size_t)Nt[t] * (HID / 4);
            normalize_agg<<<(unsigned)((n32 + 255) / 256), 256, 0, stream>>>(
                (float4*)agg[t], den[t], n32);
        }
        // ---- output projection: gelu-in + sigmoid-skip combine (WMMA) ----
        for (int t = 0; t < 2; ++t) {
            const int N = Nt[t];
            const int strips = (N + 15) / 16;
            dim3 grid((strips + 7) / 8);
            const size_t woff = (size_t)(l * 2 + t) * HID * HID;
            const size_t boff = (size_t)(l * 2 + t) * HID;
            gemm128<1,1><<<grid, 256, 0, stream>>>(agg[t], Wa + woff, ba + boff,
                hin[t], skip + l * 2 + t, opre[t], N);
        }
        // ---- gelu + layernorm ----
        float* outs[2];
        if (l == 1) { outs[0] = (float*)d_out; outs[1] = (float*)d_out + (size_t)Nu * HID; }
        else        { outs[0] = h1[0];         outs[1] = h1[1]; }
        for (int t = 0; t < 2; ++t) {
            dim3 grid((Nt[t] + 7) / 8);
            gelu_ln<<<grid, 256, 0, stream>>>(opre[t], ln_g + t * HID,
                                              ln_b + t * HID, outs[t], Nt[t]);
        }
        hin[0] = h1[0]; hin[1] = h1[1];
    }
}